// GINModel_1391569404373
// MI455X (gfx1250) — compile-verified
//
#include <hip/hip_runtime.h>
#include <math.h>

// ---------------------------------------------------------------------------
// GIN model on MI455X (gfx1250, wave32).
//   N=40000 nodes, E=640000 edges, IN=HID=128, OUT=64.
// Memory-bound (all buffers fit in 192MB L2) -> fp32 everywhere, matrix ops
// via V_WMMA_F32_16X16X4_F32 (exact fp32 matrix-core path).
// ---------------------------------------------------------------------------

typedef float v2f __attribute__((ext_vector_type(2)));
typedef float v8f __attribute__((ext_vector_type(8)));

#define N_NODES 40000
#define N_EDGES 640000
#define HID 128
#define OUT_CH 64
#define LDS_STRIDE 132   // 128 + 4 pad: 32 lanes hit 32 distinct LDS banks

// out = (1 + eps) * x  (full overwrite of the aggregation buffer)
__global__ void scale_init_kernel(const float* __restrict__ x,
                                  float* __restrict__ out,
                                  const float* __restrict__ eps,
                                  int n4) {
  int i = blockIdx.x * blockDim.x + threadIdx.x;
  if (i >= n4) return;
  float s = 1.0f + eps[0];
  float4 v = ((const float4*)x)[i];
  v.x *= s; v.y *= s; v.z *= s; v.w *= s;
  ((float4*)out)[i] = v;
}

// out[dst[e]] += feat[src[e]]  -- 32 lanes per edge, float4 per lane.
__global__ void scatter_add_kernel(const float* __restrict__ feat,
                                   const int* __restrict__ edge,  // [2, E]
                                   float* __restrict__ out,
                                   int n_edges) {
  long long tid = (long long)blockIdx.x * blockDim.x + threadIdx.x;
  int e = (int)(tid >> 5);
  if (e >= n_edges) return;
  int c = (int)(tid & 31) << 2;               // channel base (0..124)
  int s = edge[e];                            // src row
  int d = edge[n_edges + e];                  // dst row
  const float4 v = *(const float4*)(feat + (size_t)s * HID + c);
  float* o = out + (size_t)d * HID + c;
  atomicAdd(o + 0, v.x);
  atomicAdd(o + 1, v.y);
  atomicAdd(o + 2, v.z);
  atomicAdd(o + 3, v.w);
}

// out[40000x128] = relu(A[40000x128] @ W[128x128] + b)
// One block = 16 rows; 8 waves each own a 16-col tile; K loop of 32 WMMAs.
__global__ __launch_bounds__(256) void gemm128_relu_kernel(
    const float* __restrict__ A, const float* __restrict__ W,
    const float* __restrict__ bias, float* __restrict__ out) {
  __shared__ float As[16 * LDS_STRIDE];
  const int t = threadIdx.x;
  const int lane = t & 31;
  const int wave = t >> 5;                    // 0..7
  const size_t row0 = (size_t)blockIdx.x * 16;

  // Stage 16x128 A-strip into LDS: 512 float4, 256 threads x 2.
#pragma unroll
  for (int i = 0; i < 2; ++i) {
    int idx = t + i * 256;                    // float4 index
    int r = idx >> 5;
    int c = (idx & 31) << 2;
    float4 v = *(const float4*)(A + (row0 + r) * HID + c);
    float* p = &As[r * LDS_STRIDE + c];
    p[0] = v.x; p[1] = v.y; p[2] = v.z; p[3] = v.w;
  }
  __syncthreads();

  const int col = (wave << 4) + (lane & 15);  // output column 0..127
  const int kh = (lane >> 4) << 1;            // 0 (lanes 0-15) / 2 (lanes 16-31)
  const float* Arow = &As[(lane & 15) * LDS_STRIDE];

  v8f acc = {0.f, 0.f, 0.f, 0.f, 0.f, 0.f, 0.f, 0.f};
#pragma unroll
  for (int k0 = 0; k0 < HID; k0 += 4) {
    v2f a, b;
    a.x = Arow[k0 + kh];                      // A[lane%16][k0+kh+v]
    a.y = Arow[k0 + kh + 1];
    b.x = W[(size_t)(k0 + kh) * HID + col];   // W[k0+kh+v][lane%16 of tile]
    b.y = W[(size_t)(k0 + kh + 1) * HID + col];
    acc = __builtin_amdgcn_wmma_f32_16x16x4_f32(
        /*neg_a=*/false, a, /*neg_b=*/false, b,
        /*c_mod=*/(short)0, acc, /*reuse_a=*/false, /*reuse_b=*/false);
  }

  const float bv = bias[col];
  const int rhalf = (lane >> 4) << 3;         // D rows: v (+8 for lanes 16-31)
#pragma unroll
  for (int v = 0; v < 8; ++v) {
    float r = acc[v] + bv;
    r = r > 0.f ? r : 0.f;
    out[(row0 + rhalf + v) * HID + col] = r;
  }
}

// out[40000x64] = log_softmax(A[40000x128] @ W[128x64] + b), fused via LDS.
__global__ __launch_bounds__(128) void gemm64_logsoftmax_kernel(
    const float* __restrict__ A, const float* __restrict__ W,
    const float* __restrict__ bias, float* __restrict__ out) {
  __shared__ float As[16 * LDS_STRIDE];
  __shared__ float Ls[16 * 68];               // 16x64 logits, padded
  const int t = threadIdx.x;
  const int lane = t & 31;
  const int wave = t >> 5;                    // 0..3
  const size_t row0 = (size_t)blockIdx.x * 16;

#pragma unroll
  for (int i = 0; i < 4; ++i) {
    int idx = t + i * 128;
    int r = idx >> 5;
    int c = (idx & 31) << 2;
    float4 v = *(const float4*)(A + (row0 + r) * HID + c);
    float* p = &As[r * LDS_STRIDE + c];
    p[0] = v.x; p[1] = v.y; p[2] = v.z; p[3] = v.w;
  }
  __syncthreads();

  const int col = (wave << 4) + (lane & 15);  // 0..63
  const int kh = (lane >> 4) << 1;
  const float* Arow = &As[(lane & 15) * LDS_STRIDE];

  v8f acc = {0.f, 0.f, 0.f, 0.f, 0.f, 0.f, 0.f, 0.f};
#pragma unroll
  for (int k0 = 0; k0 < HID; k0 += 4) {
    v2f a, b;
    a.x = Arow[k0 + kh];
    a.y = Arow[k0 + kh + 1];
    b.x = W[(size_t)(k0 + kh) * OUT_CH + col];
    b.y = W[(size_t)(k0 + kh + 1) * OUT_CH + col];
    acc = __builtin_amdgcn_wmma_f32_16x16x4_f32(
        false, a, false, b, (short)0, acc, false, false);
  }

  const float bv = bias[col];
  const int rhalf = (lane >> 4) << 3;
#pragma unroll
  for (int v = 0; v < 8; ++v)
    Ls[(rhalf + v) * 68 + col] = acc[v] + bv;
  __syncthreads();

  if (t < 16) {                               // one thread per output row
    const float* lr = &Ls[t * 68];
    float m = lr[0];
    for (int j = 1; j < OUT_CH; ++j) m = fmaxf(m, lr[j]);
    float s = 0.f;
    for (int j = 0; j < OUT_CH; ++j) s += expf(lr[j] - m);
    const float lse = m + logf(s);
    float* orow = out + (row0 + t) * OUT_CH;
    for (int j = 0; j < OUT_CH; ++j) orow[j] = lr[j] - lse;
  }
}

extern "C" void kernel_launch(void* const* d_in, const int* in_sizes, int n_in,
                              void* d_out, int out_size, void* d_ws, size_t ws_size,
                              hipStream_t stream) {
  (void)in_sizes; (void)n_in; (void)out_size; (void)ws_size;
  const float* x    = (const float*)d_in[0];
  const int*   edge = (const int*)d_in[1];    // [2, E] int32
  const float* w1   = (const float*)d_in[2];
  const float* b1   = (const float*)d_in[3];
  const float* w2   = (const float*)d_in[4];
  const float* b2   = (const float*)d_in[5];
  const float* eps0 = (const float*)d_in[6];
  const float* w3   = (const float*)d_in[7];
  const float* b3   = (const float*)d_in[8];
  const float* w4   = (const float*)d_in[9];
  const float* b4   = (const float*)d_in[10];
  const float* eps1 = (const float*)d_in[11];

  const size_t feat_elems = (size_t)N_NODES * HID;       // 5.12M floats
  float* bufA = (float*)d_ws;                            // agg / mlp input
  float* bufB = bufA + feat_elems;                       // mlp intermediate
  float* bufC = bufB + feat_elems;                       // h (layer-0 output)

  const int n4 = (int)(feat_elems / 4);
  const int scale_blocks = (n4 + 255) / 256;
  const long long scat_threads = (long long)N_EDGES * 32;
  const int scat_blocks = (int)((scat_threads + 255) / 256);
  const int row_tiles = N_NODES / 16;                    // 2500

  // ---- layer 0 ----
  scale_init_kernel<<<scale_blocks, 256, 0, stream>>>(x, bufA, eps0, n4);
  scatter_add_kernel<<<scat_blocks, 256, 0, stream>>>(x, edge, bufA, N_EDGES);
  gemm128_relu_kernel<<<row_tiles, 256, 0, stream>>>(bufA, w1, b1, bufB);
  gemm128_relu_kernel<<<row_tiles, 256, 0, stream>>>(bufB, w2, b2, bufC); // h

  // ---- layer 1 ----
  scale_init_kernel<<<scale_blocks, 256, 0, stream>>>(bufC, bufA, eps1, n4);
  scatter_add_kernel<<<scat_blocks, 256, 0, stream>>>(bufC, edge, bufA, N_EDGES);
  gemm128_relu_kernel<<<row_tiles, 256, 0, stream>>>(bufA, w3, b3, bufB);
  gemm64_logsoftmax_kernel<<<row_tiles, 128, 0, stream>>>(bufB, w4, b4,
                                                          (float*)d_out);
}